// QuantumGNN_20203526160558
// MI455X (gfx1250) — compile-verified
//
#include <hip/hip_runtime.h>
#include <math.h>

// ---------------------------------------------------------------------------
// QuantumGNN pipeline for MI455X (gfx1250, wave32, WMMA).
//   1) H1 = relu(X @ W_e1 + b_e1)          WMMA f32 16x16x4, K=256
//   2) qf = H1 @ W_e2 + b_e2               tiny N=4, plain VALU
//   3) per-node 4-qubit statevector sim    register-resident, fully unrolled
//   4) H2 = relu(upd @ W_d1 + b_d1)        WMMA, K=4 (single WMMA per tile)
//   5) out = H2 @ W_d2 + b_d2              WMMA, K=128
// adjacency input is mathematically unused by the reference -> never read.
// ---------------------------------------------------------------------------

typedef float v2f __attribute__((ext_vector_type(2)));
typedef float v8f __attribute__((ext_vector_type(8)));

// ---------------------------------------------------------------------------
// Generic fused GEMM: C[M,N] = act(A[M,K] @ B[K,N] + bias[N])
// One wave per 16x16 output tile using V_WMMA_F32_16X16X4_F32.
// Caller guarantees M%16==0, N%16==0, K%4==0 and an exact grid (no partial
// waves), so EXEC is all-ones for every WMMA as the ISA requires.
//
// f32 WMMA VGPR layout (ISA 7.12.2):
//   A (16x4):  m = lane&15,  a.x = A[m][2*half+0], a.y = A[m][2*half+1]
//   B (4x16):  n = lane&15,  b.x = B[2*half+0][n], b.y = B[2*half+1][n]
//   C/D 16x16: acc[v] = C[v + 8*half][lane&15]
// ---------------------------------------------------------------------------
template <bool RELU>
__global__ void qgnn_wmma_gemm_bias(const float* __restrict__ A,
                                    const float* __restrict__ B,
                                    const float* __restrict__ bias,
                                    float* __restrict__ C,
                                    int M, int K, int N) {
  const int lane = threadIdx.x & 31;
  const int wave = blockIdx.x * (blockDim.x >> 5) + (threadIdx.x >> 5);
  const int tilesN = N >> 4;
  const int tm = wave / tilesN;
  const int tn = wave - tm * tilesN;

  const int half = lane >> 4;   // 0: K pair {0,1}, 1: K pair {2,3}
  const int idx  = lane & 15;   // M index for A, N index for B/C/D

  const int arow = tm * 16 + idx;
  const int bcol = tn * 16 + idx;

  const float* Ap = A + (size_t)arow * K + 2 * half;  // 8B-aligned (K even)
  const float* Bp = B + (size_t)(2 * half) * N + bcol;

  v8f acc = {0.f, 0.f, 0.f, 0.f, 0.f, 0.f, 0.f, 0.f};
  for (int k = 0; k < K; k += 4) {
    v2f a = *(const v2f*)(Ap + k);
    v2f b;
    b.x = Bp[(size_t)k * N];
    b.y = Bp[(size_t)k * N + N];
    // (neg_a, A, neg_b, B, c_mod, C, reuse_a, reuse_b)
    acc = __builtin_amdgcn_wmma_f32_16x16x4_f32(false, a, false, b,
                                                (short)0, acc, false, false);
  }

  const float bv = bias[bcol];
#pragma unroll
  for (int v = 0; v < 8; ++v) {
    const int row = tm * 16 + half * 8 + v;
    float val = acc[v] + bv;
    if (RELU) val = fmaxf(val, 0.0f);
    C[(size_t)row * N + bcol] = val;
  }
}

// ---------------------------------------------------------------------------
// qf = H1[M,128] @ W[128,4] + b[4]   (one thread per output element)
// ---------------------------------------------------------------------------
__global__ void qgnn_gemm_n4(const float* __restrict__ H1,
                             const float* __restrict__ W,
                             const float* __restrict__ b,
                             float* __restrict__ qf, int M) {
  const int t = blockIdx.x * blockDim.x + threadIdx.x;
  if (t >= M * 4) return;
  const int row = t >> 2;
  const int c = t & 3;
  float acc = b[c];
  const float* hp = H1 + (size_t)row * 128;
#pragma unroll 4
  for (int k = 0; k < 128; ++k) acc = fmaf(hp[k], W[k * 4 + c], acc);
  qf[t] = acc;
}

// ---------------------------------------------------------------------------
// Quantum statevector simulation, one thread per node, all in registers.
// ---------------------------------------------------------------------------
__device__ constexpr int qgnn_perm_chain(int b) {
  // CX(0,1); CX(1,2); CX(2,3) applied in sequence on basis bits.
  int b0 = b & 1, b1 = (b >> 1) & 1, b2 = (b >> 2) & 1, b3 = (b >> 3) & 1;
  b1 ^= b0; b2 ^= b1; b3 ^= b2;
  return b0 | (b1 << 1) | (b2 << 2) | (b3 << 3);
}
__device__ constexpr int qgnn_perm_odd(int b) {
  // CX(1,2)
  int b1 = (b >> 1) & 1;
  return b ^ ((b1 << 2) & (~b & 4) ? 4 : (b1 ? 4 : 0) & 0);  // placeholder
}

// (clean version, avoid clever bit tricks)
__device__ constexpr int qgnn_perm_odd2(int b) {
  int b1 = (b >> 1) & 1;
  int b2 = ((b >> 2) & 1) ^ b1;
  return (b & ~4) | (b2 << 2);
}

template <int QB>
__device__ __forceinline__ void qgnn_ry(float* re, float* im, float t) {
  const float c = cosf(0.5f * t);
  const float s = sinf(0.5f * t);
#pragma unroll
  for (int b = 0; b < 16; ++b) {
    if ((b & (1 << QB)) == 0) {
      const int b1 = b | (1 << QB);
      const float r0 = re[b], r1 = re[b1];
      re[b]  = c * r0 - s * r1;
      re[b1] = s * r0 + c * r1;
      const float i0 = im[b], i1 = im[b1];
      im[b]  = c * i0 - s * i1;
      im[b1] = s * i0 + c * i1;
    }
  }
}

__device__ __forceinline__ void qgnn_rz(float* re, float* im,
                                        float t0, float t1, float t2, float t3) {
#pragma unroll
  for (int b = 0; b < 16; ++b) {
    const float ang = 0.5f * (((b & 1) ? t0 : -t0) + ((b & 2) ? t1 : -t1) +
                              ((b & 4) ? t2 : -t2) + ((b & 8) ? t3 : -t3));
    const float cs = cosf(ang), sn = sinf(ang);
    const float r = re[b], i = im[b];
    re[b] = r * cs - i * sn;
    im[b] = r * sn + i * cs;
  }
}

__global__ void qgnn_quantum(const float* __restrict__ qf,
                             const float* __restrict__ theta,  // 24 params
                             float* __restrict__ upd, int N) {
  const int node = blockIdx.x * blockDim.x + threadIdx.x;
  if (node >= N) return;

  float q[4];
  float n1 = 0.f;
#pragma unroll
  for (int i = 0; i < 4; ++i) { q[i] = qf[node * 4 + i]; n1 += q[i] * q[i]; }
  n1 = sqrtf(n1);
#pragma unroll
  for (int i = 0; i < 4; ++i) q[i] /= (n1 + 1e-8f);
  float n2 = 0.f;
#pragma unroll
  for (int i = 0; i < 4; ++i) n2 += q[i] * q[i];
  n2 = sqrtf(n2);
#pragma unroll
  for (int i = 0; i < 4; ++i) q[i] /= (n2 + 1e-8f);
  float n3 = 0.f;
#pragma unroll
  for (int i = 0; i < 4; ++i) n3 += q[i] * q[i];
  n3 = sqrtf(n3);

  float re[16], im[16];
#pragma unroll
  for (int b = 0; b < 16; ++b) { re[b] = 0.f; im[b] = 0.f; }
#pragma unroll
  for (int i = 0; i < 4; ++i) re[i] = q[i] / n3;

#pragma unroll
  for (int layer = 0; layer < 2; ++layer) {
    const float* th = theta + layer * 12;
    // RY stage A: theta[3i] on qubit i
    qgnn_ry<0>(re, im, th[0]);
    qgnn_ry<1>(re, im, th[3]);
    qgnn_ry<2>(re, im, th[6]);
    qgnn_ry<3>(re, im, th[9]);
    // CX chain permutation: new[p(b)] = old[b]
    {
      float tr[16], ti[16];
#pragma unroll
      for (int b = 0; b < 16; ++b) {
        const int p = qgnn_perm_chain(b);
        tr[p] = re[b]; ti[p] = im[b];
      }
#pragma unroll
      for (int b = 0; b < 16; ++b) { re[b] = tr[b]; im[b] = ti[b]; }
    }
    // RZ stage: theta[3i+1]
    qgnn_rz(re, im, th[1], th[4], th[7], th[10]);
    // CX odd permutation
    {
      float tr[16], ti[16];
#pragma unroll
      for (int b = 0; b < 16; ++b) {
        const int p = qgnn_perm_odd2(b);
        tr[p] = re[b]; ti[p] = im[b];
      }
#pragma unroll
      for (int b = 0; b < 16; ++b) { re[b] = tr[b]; im[b] = ti[b]; }
    }
    // RY stage C: theta[3i+2]
    qgnn_ry<0>(re, im, th[2]);
    qgnn_ry<1>(re, im, th[5]);
    qgnn_ry<2>(re, im, th[8]);
    qgnn_ry<3>(re, im, th[11]);
  }

#pragma unroll
  for (int i = 0; i < 4; ++i) upd[node * 4 + i] = re[i];
}

// ---------------------------------------------------------------------------
// Launch
// ---------------------------------------------------------------------------
extern "C" void kernel_launch(void* const* d_in, const int* in_sizes, int n_in,
                              void* d_out, int out_size, void* d_ws, size_t ws_size,
                              hipStream_t stream) {
  const float* X     = (const float*)d_in[0];   // (10000, 256)
  // d_in[1] = adjacency: unused by the reference (400 MB we never touch)
  const float* W_e1  = (const float*)d_in[2];   // (256, 128)
  const float* b_e1  = (const float*)d_in[3];   // (128,)
  const float* W_e2  = (const float*)d_in[4];   // (128, 4)
  const float* b_e2  = (const float*)d_in[5];   // (4,)
  const float* mp    = (const float*)d_in[6];   // (24,)
  const float* W_d1  = (const float*)d_in[7];   // (4, 128)
  const float* b_d1  = (const float*)d_in[8];   // (128,)
  const float* W_d2  = (const float*)d_in[9];   // (128, 128)
  const float* b_d2  = (const float*)d_in[10];  // (128,)
  float* out = (float*)d_out;                   // (10000, 128)

  const int M = in_sizes[0] / 256;  // 10000
  const int DIN = 256, HID = 128;

  float* ws = (float*)d_ws;
  float* H1  = ws;                          // M*128
  float* qf  = H1 + (size_t)M * HID;        // M*4
  float* upd = qf + (size_t)M * 4;          // M*4
  float* H2  = upd + (size_t)M * 4;         // M*128

  const int WPB = 4;                 // waves per block (128 threads)
  const int tiles = (M / 16) * (HID / 16);   // 625*8 = 5000, exact
  const int gemmBlocks = tiles / WPB;        // 1250, exact

  // 1) H1 = relu(X @ W_e1 + b_e1)
  qgnn_wmma_gemm_bias<true><<<gemmBlocks, WPB * 32, 0, stream>>>(
      X, W_e1, b_e1, H1, M, DIN, HID);

  // 2) qf = H1 @ W_e2 + b_e2
  qgnn_gemm_n4<<<(M * 4 + 255) / 256, 256, 0, stream>>>(H1, W_e2, b_e2, qf, M);

  // 3) quantum message passing (register-resident statevector sim)
  qgnn_quantum<<<(M + 255) / 256, 256, 0, stream>>>(qf, mp, upd, M);

  // 4) H2 = relu(upd @ W_d1 + b_d1)   (K=4 -> one WMMA per tile)
  qgnn_wmma_gemm_bias<true><<<gemmBlocks, WPB * 32, 0, stream>>>(
      upd, W_d1, b_d1, H2, M, 4, HID);

  // 5) out = H2 @ W_d2 + b_d2
  qgnn_wmma_gemm_bias<false><<<gemmBlocks, WPB * 32, 0, stream>>>(
      H2, W_d2, b_d2, out, M, HID, HID);
}